// TripletPromptEncoder_15642270892541
// MI455X (gfx1250) — compile-verified
//
#include <hip/hip_runtime.h>
#include <stdint.h>

// TripletPromptEncoder: B=32768, D=512, V=32000, MAX_TOK=5.
// Memory-bound scatter/gather: ~400 MB compulsory traffic -> ~17 us floor
// at 23.3 TB/s. One wave (32 lanes) per sample; each lane owns 4x float4
// chunks of a D=512 row. Grid-stride over samples to keep LDS-staging
// re-reads (14 KB/block) negligible.

#define DDIM   512
#define ROWS   5
#define CHUNKS (DDIM / 4 / 32)   // 4 float4 per lane per row
#define WAVES_PER_BLOCK 8        // 256 threads
#define MAX_BLOCKS 1024

typedef __attribute__((ext_vector_type(4))) float v4f;

__device__ __forceinline__ uint32_t lds_byte_off(const void* p) {
  // Generic pointers into LDS carry the wave-relative LDS offset in the low
  // 32 bits (aperture base lives in the high bits) -- truncate to get dsaddr.
  return (uint32_t)(uintptr_t)p;
}

__device__ __forceinline__ void async_copy_b128(uint32_t lds_addr, const void* gaddr) {
  // gfx1250 async DMA: memory -> LDS, tracked by ASYNCcnt (CDNA5 ISA op 98).
  asm volatile("global_load_async_to_lds_b128 %0, %1, off"
               :: "v"(lds_addr), "v"(gaddr)
               : "memory");
}

__device__ __forceinline__ void wait_async0() {
  asm volatile("s_wait_asynccnt 0" ::: "memory");
}

__device__ __forceinline__ void nt_store(v4f* p, v4f v) {
  // Output is 335 MB streamed and never re-read: keep it out of L2 (TH=NT).
  __builtin_nontemporal_store(v, p);
}

__global__ __launch_bounds__(256, 4)
void triplet_prompt_encoder_kernel(
    const float*   __restrict__ code_table,       // [V, D]
    const float*   __restrict__ date_w,           // [D]
    const float*   __restrict__ date_b,           // [D]
    const float*   __restrict__ val_w,            // [D]
    const float*   __restrict__ val_b,            // [D]
    const float*   __restrict__ ts_token,         // [D]
    const float*   __restrict__ code_prefix,      // [D]
    const float*   __restrict__ val_prefix,       // [D]
    const float*   __restrict__ numerical_value,  // [B]
    const float*   __restrict__ time_delta_days,  // [B]
    const int*     __restrict__ code,             // [B]
    const uint8_t* __restrict__ static_mask,      // [B] bool
    const uint8_t* __restrict__ num_mask,         // [B] bool
    float*         __restrict__ out,              // [B, 5, D]
    int Bn)
{
  // 7 parameter vectors (14 KB) staged once per workgroup via async DMA.
  // Compile-time array selection: no dynamic pointer cndmask chains.
  __shared__ __align__(16) float s_par[7 * DDIM];
  {
    const int e = threadIdx.x & 127;                 // 16B chunk index 0..127
    const uint32_t lbase = lds_byte_off(s_par) + (uint32_t)e * 16u;
    if (threadIdx.x < 128) {                         // waves 0..3
      async_copy_b128(lbase + 0u * 2048u, (const v4f*)date_w     + e);
      async_copy_b128(lbase + 2u * 2048u, (const v4f*)val_w      + e);
      async_copy_b128(lbase + 4u * 2048u, (const v4f*)ts_token   + e);
      async_copy_b128(lbase + 6u * 2048u, (const v4f*)val_prefix + e);
    } else {                                         // waves 4..7
      async_copy_b128(lbase + 1u * 2048u, (const v4f*)date_b      + e);
      async_copy_b128(lbase + 3u * 2048u, (const v4f*)val_b       + e);
      async_copy_b128(lbase + 5u * 2048u, (const v4f*)code_prefix + e);
    }
    wait_async0();       // own wave's async copies landed in LDS
    __syncthreads();     // make all waves' chunks visible
  }

  const v4f* S  = (const v4f*)s_par;
  const v4f* DW = S;           // date_w
  const v4f* DB = S + 128;     // date_b
  const v4f* VW = S + 256;     // val_w
  const v4f* VB = S + 384;     // val_b
  const v4f* TS = S + 512;     // ts_token
  const v4f* CP = S + 640;     // code_prefix
  const v4f* VP = S + 768;     // val_prefix

  const int wave   = threadIdx.x >> 5;
  const int lane   = threadIdx.x & 31;
  const int stride = gridDim.x * WAVES_PER_BLOCK;

  for (int b = blockIdx.x * WAVES_PER_BLOCK + wave; b < Bn; b += stride) {
    const float t  = time_delta_days[b];
    const float nv = numerical_value[b];
    const bool  tv = (t != 0.0f);            // time_valid_mask
    const bool  vv = (num_mask[b] != 0);     // numerical_value_mask
    const bool  sm = (static_mask[b] != 0);
    const int   c  = code[b];

    const v4f* crow = (const v4f*)code_table + (size_t)c * (DDIM / 4);
    // Warm the 2 KB gather row while the LDS-sourced rows are emitted.
    __builtin_prefetch((const char*)crow + lane * 64, 0, 0);

    v4f* orow = (v4f*)out + (size_t)b * (ROWS * (DDIM / 4));

    // Compacted slot order: [ts?, code_prefix, code_emb, val_prefix?, val_emb?]
    int r = 0;

    if (tv) {  // ts slot present only when time_delta != 0
#pragma unroll
      for (int j = 0; j < CHUNKS; ++j) {
        const int i = lane + 32 * j;
        v4f v;
        if (sm) v = t * DW[i] + DB[i];   // time_emb (static & t_valid)
        else    v = TS[i];               // ts_token
        nt_store(&orow[i], v);
      }
      r = 1;
    }

    {  // code_prefix (always present)
      v4f* dst = orow + r * 128;
#pragma unroll
      for (int j = 0; j < CHUNKS; ++j) {
        const int i = lane + 32 * j;
        nt_store(&dst[i], CP[i]);
      }
      ++r;
    }

    {  // code embedding gather (always present)
      v4f* dst = orow + r * 128;
#pragma unroll
      for (int j = 0; j < CHUNKS; ++j) {
        const int i = lane + 32 * j;
        nt_store(&dst[i], crow[i]);
      }
      ++r;
    }

    if (vv) {  // val_prefix + val_emb
      v4f* dst = orow + r * 128;
#pragma unroll
      for (int j = 0; j < CHUNKS; ++j) {
        const int i = lane + 32 * j;
        nt_store(&dst[i], VP[i]);
      }
      ++r;
      dst = orow + r * 128;
#pragma unroll
      for (int j = 0; j < CHUNKS; ++j) {
        const int i = lane + 32 * j;
        nt_store(&dst[i], nv * VW[i] + VB[i]);
      }
      ++r;
    }

    // Zero-pad remaining slots (reference output is zero-padded ragged buffer).
    const v4f z = {0.0f, 0.0f, 0.0f, 0.0f};
    for (; r < ROWS; ++r) {
      v4f* dst = orow + r * 128;
#pragma unroll
      for (int j = 0; j < CHUNKS; ++j) {
        nt_store(&dst[lane + 32 * j], z);
      }
    }
  }
}

extern "C" void kernel_launch(void* const* d_in, const int* in_sizes, int n_in,
                              void* d_out, int out_size, void* d_ws, size_t ws_size,
                              hipStream_t stream) {
  const float*   code_table      = (const float*)d_in[0];
  const float*   date_w          = (const float*)d_in[1];
  const float*   date_b          = (const float*)d_in[2];
  const float*   val_w           = (const float*)d_in[3];
  const float*   val_b           = (const float*)d_in[4];
  const float*   ts_token        = (const float*)d_in[5];
  const float*   code_prefix     = (const float*)d_in[6];
  const float*   val_prefix      = (const float*)d_in[7];
  const float*   numerical_value = (const float*)d_in[8];
  const float*   time_delta_days = (const float*)d_in[9];
  const int*     code            = (const int*)d_in[10];
  const uint8_t* static_mask     = (const uint8_t*)d_in[11];  // jax bool = 1 byte
  const uint8_t* num_mask        = (const uint8_t*)d_in[12];  // jax bool = 1 byte

  const int Bn = in_sizes[8];  // numerical_value length == B
  int blocks = (Bn + WAVES_PER_BLOCK - 1) / WAVES_PER_BLOCK;
  if (blocks > MAX_BLOCKS) blocks = MAX_BLOCKS;
  if (blocks < 1) blocks = 1;

  hipLaunchKernelGGL(triplet_prompt_encoder_kernel, dim3(blocks), dim3(256), 0,
                     stream, code_table, date_w, date_b, val_w, val_b, ts_token,
                     code_prefix, val_prefix, numerical_value, time_delta_days,
                     code, static_mask, num_mask, (float*)d_out, Bn);
}